// DHCN_87531433493067
// MI455X (gfx1250) — compile-verified
//
#include <hip/hip_runtime.h>

// ---------------------------------------------------------------------------
// DHCN pipeline for MI455X (gfx1250, wave32, WMMA + double-buffered async LDS)
//   out0: item_emb    [50000,128]  (HyperConv: 2x COO SpMM + sum)
//   out1: session_emb [1024,128]   (LineConv: pool + (D@A) propagation)
//   out2: hy          [1024,50,128](GNN: gathered einsum + GRU gates)
// ---------------------------------------------------------------------------

typedef __attribute__((ext_vector_type(2))) float v2f;
typedef __attribute__((ext_vector_type(8))) float v8f;

#define N_NODE 50000
#define EMB    128
#define BATCH  1024
#define SLEN   50
#define NEDGE  1600000

// Toggle: stage GEMM tiles with gfx1250 async global->LDS copies (ASYNCcnt).
#define USE_ASYNC_LDS 1

// ---------------------------------------------------------------------------
// gfx1250 async global->LDS copy helpers (per-lane LDS byte addr + global addr)
// ---------------------------------------------------------------------------
__device__ __forceinline__ void async_g2l_b128(unsigned lds_addr, const float* gptr) {
    unsigned long long ga = (unsigned long long)(uintptr_t)gptr;
    asm volatile("global_load_async_to_lds_b128 %0, %1, off"
                 :: "v"(lds_addr), "v"(ga) : "memory");
}
__device__ __forceinline__ void wait_async_all() {
    asm volatile("s_wait_asynccnt 0x0" ::: "memory");
}

// ---------------------------------------------------------------------------
// Elementwise helpers
// ---------------------------------------------------------------------------
__global__ __launch_bounds__(256) void zero_f32(float* __restrict__ p, int n) {
    int i = blockIdx.x * blockDim.x + threadIdx.x;
    if (i < n) p[i] = 0.0f;
}

__global__ __launch_bounds__(256) void add3_f32(const float* __restrict__ a,
                                                const float* __restrict__ b,
                                                const float* __restrict__ c,
                                                float* __restrict__ o, int n) {
    int i = blockIdx.x * blockDim.x + threadIdx.x;
    if (i < n) o[i] = a[i] + b[i] + c[i];
}

// ---------------------------------------------------------------------------
// HyperConv SpMM layer: y[row] += val * x[col]   (COO, atomic scatter)
// one wave32 per edge; 128 floats/row = float4 per lane
// ---------------------------------------------------------------------------
__global__ __launch_bounds__(256)
void spmm_coo_atomic(const int* __restrict__ rows, const int* __restrict__ cols,
                     const float* __restrict__ vals, const float* __restrict__ x,
                     float* __restrict__ y, int E) {
    int wid  = (blockIdx.x * blockDim.x + threadIdx.x) >> 5;
    int lane = threadIdx.x & 31;
    if (wid >= E) return;
    int   r = rows[wid];
    int   c = cols[wid];
    float v = vals[wid];
    const float4 xv = *(const float4*)(x + (size_t)c * EMB + lane * 4);
    float* yp = y + (size_t)r * EMB + lane * 4;
    atomicAdd(yp + 0, v * xv.x);
    atomicAdd(yp + 1, v * xv.y);
    atomicAdd(yp + 2, v * xv.z);
    atomicAdd(yp + 3, v * xv.w);
}

// ---------------------------------------------------------------------------
// Session mean pool with zero-pad row: sess[b,d] = sum_l pad(emb)[item[b,l]][d] / len
// ---------------------------------------------------------------------------
__global__ __launch_bounds__(128)
void sess_pool(const int* __restrict__ items, const float* __restrict__ emb,
               const float* __restrict__ slen, float* __restrict__ sess) {
    int b = blockIdx.x, d = threadIdx.x;
    float s = 0.0f;
    for (int l = 0; l < SLEN; ++l) {
        int idx = items[b * SLEN + l];
        if (idx > 0) s += emb[(size_t)(idx - 1) * EMB + d];
    }
    sess[(size_t)b * EMB + d] = s / slen[b];
}

// ---------------------------------------------------------------------------
// Tiled f32 WMMA GEMM:  C[M,N] = A[M,K] @ op(B) (+ bias[n])
//   TRANSB==0: op(B)[k][n] = B[k*ldb + n]   (LDS tile stored k-major [16][64])
//   TRANSB==1: op(B)[k][n] = B[n*ldb + k]   (LDS tile stored n-major [64][16])
// Block tile 128x64, 8 waves; each wave: 2x2 tiles of 16x16 via
// V_WMMA_F32_16X16X4_F32. K staged in 16-wide chunks, DOUBLE-BUFFERED through
// LDS with GLOBAL_LOAD_ASYNC_TO_LDS_B128 (ASYNCcnt): chunk k+1 DMA overlaps
// chunk k WMMA compute.
// Requires M%128==0, N%64==0, K%16==0 (true for every GEMM in this pipeline).
// ---------------------------------------------------------------------------
#define BUF_FLOATS 3072          // 128*16 (A) + 64*16 (B) floats per buffer
#define BUF_BYTES  12288
#define LDSB_OFF_BYTES 8192      // byte offset of B tile within a buffer
#define LDSB_OFF_FLOATS 2048

template <int TRANSB>
__global__ __launch_bounds__(256)
void wmma_gemm_f32(const float* __restrict__ A, int lda,
                   const float* __restrict__ B, int ldb,
                   float* __restrict__ C, int ldc,
                   const float* __restrict__ bias,
                   int M, int N, int K) {
    // single shared array => buffer 0 at group-segment byte offset 0
    __shared__ float lds[2 * BUF_FLOATS];

    const int tid   = threadIdx.x;
    const int wave  = tid >> 5;
    const int lane  = tid & 31;
    const int wm    = wave >> 1;          // 0..3  (M direction)
    const int wn    = wave & 1;           // 0..1  (N direction)
    const int bm    = blockIdx.y * 128;
    const int bn    = blockIdx.x * 64;
    const int lhalf = lane >> 4;          // 0/1 lane half
    const int l15   = lane & 15;

    // stage one 16-wide K chunk into LDS buffer `buf` (3x async b128 / thread)
    auto stage = [&](int k0, int buf) {
        unsigned base = (unsigned)buf * BUF_BYTES;
        // A tile 128x16, row-major (512 float4 / 256 threads = 2 each)
        #pragma unroll
        for (int it = 0; it < 2; ++it) {
            int g   = tid + it * 256;
            int row = g >> 2;
            int c4  = (g & 3) * 4;
            const float* ga = A + (size_t)(bm + row) * lda + k0 + c4;
#if USE_ASYNC_LDS
            async_g2l_b128(base + (unsigned)(row * 16 + c4) * 4u, ga);
#else
            *(float4*)(lds + buf * BUF_FLOATS + row * 16 + c4) = *(const float4*)ga;
#endif
        }
        if (TRANSB == 0) {
            // B tile 16x64 k-major
            int row = tid >> 4;              // k within chunk
            int c4  = (tid & 15) * 4;        // n within tile
            const float* gb = B + (size_t)(k0 + row) * ldb + bn + c4;
#if USE_ASYNC_LDS
            async_g2l_b128(base + LDSB_OFF_BYTES + (unsigned)(row * 64 + c4) * 4u, gb);
#else
            *(float4*)(lds + buf * BUF_FLOATS + LDSB_OFF_FLOATS + row * 64 + c4) =
                *(const float4*)gb;
#endif
        } else {
            // B tile stored n-major [64][16]; transpose absorbed at read time
            int n  = tid >> 2;               // 0..63
            int k4 = (tid & 3) * 4;          // 0,4,8,12
            const float* gb = B + (size_t)(bn + n) * ldb + k0 + k4;
#if USE_ASYNC_LDS
            async_g2l_b128(base + LDSB_OFF_BYTES + (unsigned)(n * 16 + k4) * 4u, gb);
#else
            *(float4*)(lds + buf * BUF_FLOATS + LDSB_OFF_FLOATS + n * 16 + k4) =
                *(const float4*)gb;
#endif
        }
    };

    v8f acc[2][2] = {};
    const int nch = K >> 4;

    stage(0, 0);                              // prologue: chunk 0 -> buffer 0

    for (int ch = 0; ch < nch; ++ch) {
        const int cur = ch & 1;
        const float* lA = lds + cur * BUF_FLOATS;
        const float* lB = lA + LDSB_OFF_FLOATS;

#if USE_ASYNC_LDS
        wait_async_all();                     // my async writes (cur buf) done
#endif
        __syncthreads();                      // everyone's writes visible;
                                              // also fences prior-chunk reads
        if (ch + 1 < nch) stage((ch + 1) * 16, cur ^ 1);   // overlap DMA

        // ---- consume chunk: 4 k-steps x (2x2) WMMA ----
        #pragma unroll
        for (int kk = 0; kk < 16; kk += 4) {
            v2f af[2], bf[2];
            #pragma unroll
            for (int i = 0; i < 2; ++i) {
                // A 16x4 fragment: lanes 0-15 K=kk+0/+1, lanes 16-31 K=kk+2/+3
                int arow = wm * 32 + i * 16 + l15;
                af[i].x = lA[arow * 16 + kk + lhalf * 2 + 0];
                af[i].y = lA[arow * 16 + kk + lhalf * 2 + 1];
            }
            #pragma unroll
            for (int j = 0; j < 2; ++j) {
                int bcol = wn * 32 + j * 16 + l15;
                if (TRANSB == 0) {
                    bf[j].x = lB[(kk + lhalf * 2 + 0) * 64 + bcol];
                    bf[j].y = lB[(kk + lhalf * 2 + 1) * 64 + bcol];
                } else {
                    // n-major tile: k-pair contiguous -> single ds_load_b64
                    bf[j].x = lB[bcol * 16 + kk + lhalf * 2 + 0];
                    bf[j].y = lB[bcol * 16 + kk + lhalf * 2 + 1];
                }
            }
            #pragma unroll
            for (int i = 0; i < 2; ++i)
                #pragma unroll
                for (int j = 0; j < 2; ++j)
                    acc[i][j] = __builtin_amdgcn_wmma_f32_16x16x4_f32(
                        /*neg_a=*/false, af[i], /*neg_b=*/false, bf[j],
                        /*c_mod=*/(short)0, acc[i][j],
                        /*reuse_a=*/false, /*reuse_b=*/false);
        }
    }

    // ---- store (C 16x16 layout: VGPR r -> M=r(+8 for lane half), N=l15) ----
    #pragma unroll
    for (int i = 0; i < 2; ++i) {
        #pragma unroll
        for (int j = 0; j < 2; ++j) {
            int row0 = bm + wm * 32 + i * 16 + lhalf * 8;
            int col  = bn + wn * 32 + j * 16 + l15;
            float bv = bias ? bias[col] : 0.0f;
            #pragma unroll
            for (int r = 0; r < 8; ++r) {
                C[(size_t)(row0 + r) * ldc + col] = acc[i][j][r] + bv;
            }
        }
    }
}

// ---------------------------------------------------------------------------
// GNN gathered einsum:
//   inputs[b,i, 0:128]  = sum_j A_gnn[b, p,   j] * h_in [b,j,:] + b_iah
//   inputs[b,i,128:256] = sum_j A_gnn[b, p, L+j] * h_out[b,j,:] + b_oah
// where p = alias_pos[b,i]
// ---------------------------------------------------------------------------
__global__ __launch_bounds__(128)
void gnn_einsum(const float* __restrict__ A_gnn, const int* __restrict__ alias_pos,
                const float* __restrict__ h_in, const float* __restrict__ h_out,
                const float* __restrict__ b_iah, const float* __restrict__ b_oah,
                float* __restrict__ inputs) {
    int i = blockIdx.x;   // 0..49
    int b = blockIdx.y;   // 0..1023
    int d = threadIdx.x;  // 0..127
    int p = alias_pos[b * SLEN + i];
    const float* arow = A_gnn + ((size_t)b * SLEN + p) * (2 * SLEN);
    const float* hin  = h_in  + (size_t)b * SLEN * EMB + d;
    const float* hou  = h_out + (size_t)b * SLEN * EMB + d;
    float s_in = 0.0f, s_out = 0.0f;
    #pragma unroll 5
    for (int j = 0; j < SLEN; ++j) {
        s_in  += arow[j]        * hin[(size_t)j * EMB];
        s_out += arow[SLEN + j] * hou[(size_t)j * EMB];
    }
    float* op = inputs + ((size_t)b * SLEN + i) * (2 * EMB);
    op[d]       = s_in  + b_iah[d];
    op[EMB + d] = s_out + b_oah[d];
}

// ---------------------------------------------------------------------------
// Fused GRU gates:
//   r = sig(i_r+h_r); z = sig(i_i+h_i); n = tanh(i_n + r*h_n)
//   hy = n + z*(h - n)
// ---------------------------------------------------------------------------
__global__ __launch_bounds__(256)
void gru_fuse(const float* __restrict__ gi, const float* __restrict__ gh,
              const float* __restrict__ hidden, float* __restrict__ hy, int n) {
    int id = blockIdx.x * blockDim.x + threadIdx.x;
    if (id >= n) return;
    int row = id >> 7;     // b*L + l
    int d   = id & 127;
    size_t base = (size_t)row * (3 * EMB);
    float i_r = gi[base + d], i_i = gi[base + EMB + d], i_n = gi[base + 2 * EMB + d];
    float h_r = gh[base + d], h_i = gh[base + EMB + d], h_n = gh[base + 2 * EMB + d];
    float rg = 1.0f / (1.0f + __expf(-(i_r + h_r)));
    float zg = 1.0f / (1.0f + __expf(-(i_i + h_i)));
    float ng = tanhf(i_n + rg * h_n);
    float h  = hidden[id];
    hy[id] = ng + zg * (h - ng);
}

// ---------------------------------------------------------------------------
// Orchestration
// ---------------------------------------------------------------------------
extern "C" void kernel_launch(void* const* d_in, const int* in_sizes, int n_in,
                              void* d_out, int out_size, void* d_ws, size_t ws_size,
                              hipStream_t stream) {
    const int*   adj_row   = (const int*)  d_in[0];
    const int*   adj_col   = (const int*)  d_in[1];
    const float* adj_val   = (const float*)d_in[2];
    const float* embedding = (const float*)d_in[3];
    const int*   sess_item = (const int*)  d_in[4];
    const float* sess_len  = (const float*)d_in[5];
    const float* D         = (const float*)d_in[6];
    const float* A_sess    = (const float*)d_in[7];
    const float* A_gnn     = (const float*)d_in[8];
    const int*   alias_pos = (const int*)  d_in[9];
    const float* hidden    = (const float*)d_in[10];
    const float* W_in      = (const float*)d_in[11];
    const float* b_in      = (const float*)d_in[12];
    const float* W_out     = (const float*)d_in[13];
    const float* b_out     = (const float*)d_in[14];
    const float* w_ih      = (const float*)d_in[15];
    const float* b_ih      = (const float*)d_in[16];
    const float* w_hh      = (const float*)d_in[17];
    const float* b_hh      = (const float*)d_in[18];
    const float* b_iah     = (const float*)d_in[19];
    const float* b_oah     = (const float*)d_in[20];

    float* out_item = (float*)d_out;                               // 6,400,000
    float* out_sess = (float*)d_out + (size_t)N_NODE * EMB;        //   131,072
    float* out_hy   = out_sess + (size_t)BATCH * EMB;              // 6,553,600

    // workspace layout (floats)
    float* ws = (float*)d_ws;
    size_t o = 0;
    const size_t NE   = (size_t)N_NODE * EMB;          // 6,400,000
    const size_t BL   = (size_t)BATCH * SLEN;          // 51,200
    float* emb1   = ws + o; o += NE;
    float* emb2   = ws + o; o += NE;
    float* sess   = ws + o; o += (size_t)BATCH * EMB;
    float* DA     = ws + o; o += (size_t)BATCH * BATCH;
    float* s1     = ws + o; o += (size_t)BATCH * EMB;
    float* s2     = ws + o; o += (size_t)BATCH * EMB;
    float* h_in   = ws + o; o += BL * EMB;
    float* h_out  = ws + o; o += BL * EMB;
    float* inputs = ws + o; o += BL * 2 * EMB;
    float* gi     = ws + o; o += BL * 3 * EMB;
    float* gh     = ws + o; o += BL * 3 * EMB;
    (void)o; (void)ws_size; (void)in_sizes; (void)n_in; (void)out_size;

    // ---- Phase 1: HyperConv (SpMM x2, atomic scatter) ----
    zero_f32<<<dim3((NE + 255) / 256), dim3(256), 0, stream>>>(emb1, (int)NE);
    zero_f32<<<dim3((NE + 255) / 256), dim3(256), 0, stream>>>(emb2, (int)NE);
    spmm_coo_atomic<<<dim3(NEDGE / 8), dim3(256), 0, stream>>>(
        adj_row, adj_col, adj_val, embedding, emb1, NEDGE);
    spmm_coo_atomic<<<dim3(NEDGE / 8), dim3(256), 0, stream>>>(
        adj_row, adj_col, adj_val, emb1, emb2, NEDGE);
    add3_f32<<<dim3((NE + 255) / 256), dim3(256), 0, stream>>>(
        embedding, emb1, emb2, out_item, (int)NE);

    // ---- Phase 2: LineConv ----
    sess_pool<<<dim3(BATCH), dim3(EMB), 0, stream>>>(sess_item, embedding, sess_len, sess);
    // DA = D @ A_sess   [1024,1024,1024]
    wmma_gemm_f32<0><<<dim3(BATCH / 64, BATCH / 128), dim3(256), 0, stream>>>(
        D, BATCH, A_sess, BATCH, DA, BATCH, nullptr, BATCH, BATCH, BATCH);
    // s1 = DA @ sess ; s2 = DA @ s1
    wmma_gemm_f32<0><<<dim3(EMB / 64, BATCH / 128), dim3(256), 0, stream>>>(
        DA, BATCH, sess, EMB, s1, EMB, nullptr, BATCH, EMB, BATCH);
    wmma_gemm_f32<0><<<dim3(EMB / 64, BATCH / 128), dim3(256), 0, stream>>>(
        DA, BATCH, s1, EMB, s2, EMB, nullptr, BATCH, EMB, BATCH);
    add3_f32<<<dim3((BATCH * EMB + 255) / 256), dim3(256), 0, stream>>>(
        sess, s1, s2, out_sess, BATCH * EMB);

    // ---- Phase 3: GNN ----
    // h_in = hidden @ W_in.T + b_in ; h_out = hidden @ W_out.T + b_out
    wmma_gemm_f32<1><<<dim3(EMB / 64, BL / 128), dim3(256), 0, stream>>>(
        hidden, EMB, W_in, EMB, h_in, EMB, b_in, (int)BL, EMB, EMB);
    wmma_gemm_f32<1><<<dim3(EMB / 64, BL / 128), dim3(256), 0, stream>>>(
        hidden, EMB, W_out, EMB, h_out, EMB, b_out, (int)BL, EMB, EMB);
    // gathered per-session einsums -> inputs [BL, 256]
    gnn_einsum<<<dim3(SLEN, BATCH), dim3(EMB), 0, stream>>>(
        A_gnn, alias_pos, h_in, h_out, b_iah, b_oah, inputs);
    // gi = inputs @ w_ih.T + b_ih   [51200, 384, 256]
    wmma_gemm_f32<1><<<dim3(3 * EMB / 64, BL / 128), dim3(256), 0, stream>>>(
        inputs, 2 * EMB, w_ih, 2 * EMB, gi, 3 * EMB, b_ih, (int)BL, 3 * EMB, 2 * EMB);
    // gh = hidden @ w_hh.T + b_hh   [51200, 384, 128]
    wmma_gemm_f32<1><<<dim3(3 * EMB / 64, BL / 128), dim3(256), 0, stream>>>(
        hidden, EMB, w_hh, EMB, gh, 3 * EMB, b_hh, (int)BL, 3 * EMB, EMB);
    // fused GRU gates -> hy
    int ngru = (int)(BL * EMB);
    gru_fuse<<<dim3((ngru + 255) / 256), dim3(256), 0, stream>>>(gi, gh, hidden, out_hy, ngru);
}